// GRIDS_63213328662785
// MI455X (gfx1250) — compile-verified
//
#include <hip/hip_runtime.h>
#include <math.h>

#define B_ 8
#define C_ 96
#define N_ 3136
#define K_ 18

typedef __attribute__((ext_vector_type(2))) float v2f;
typedef __attribute__((ext_vector_type(8))) float v8f;

// ---------------------------------------------------------------------------
// Tiled transpose: in [b][R][Ccols] -> out [b][Ccols][R]   (blockIdx.z = batch)
// ---------------------------------------------------------------------------
__global__ __launch_bounds__(256) void transpose_k(const float* __restrict__ in,
                                                   float* __restrict__ out,
                                                   int R, int Ccols)
{
    __shared__ float tile[32][33];
    const int b = blockIdx.z;
    const float* ip = in  + (size_t)b * R * Ccols;
    float*       op = out + (size_t)b * R * Ccols;
    const int c0 = blockIdx.x * 32;
    const int r0 = blockIdx.y * 32;
    const int tx = threadIdx.x, ty = threadIdx.y;
#pragma unroll
    for (int i = 0; i < 32; i += 8) {
        int r = r0 + ty + i, c = c0 + tx;
        tile[ty + i][tx] = (r < R && c < Ccols) ? ip[(size_t)r * Ccols + c] : 0.0f;
    }
    __syncthreads();
#pragma unroll
    for (int i = 0; i < 32; i += 8) {
        int r = c0 + ty + i;   // output row = input column
        int c = r0 + tx;       // output col = input row
        if (r < Ccols && c < R) op[(size_t)r * R + c] = tile[tx][ty + i];
    }
}

// ---------------------------------------------------------------------------
// Main kernel: one wave32 per (b, n).
// Channel statistics via FP32 Gram matrix on V_WMMA_F32_16X16X4_F32:
//   16-row group = [x_i rows for 8 edges | x_j rows for the same 8 edges],
//   D += A * A^T over 24 chunks of K=4 channels (A passed as both operands —
//   the f32 A and B lane layouts coincide under transpose).
//   D[e][e]      = sum x_i^2   -> acc[e] @ lane e
//   D[e][8+e]    = sum x_i*x_j -> acc[e] @ lane 8+e
//   D[8+e][8+e]  = sum x_j^2   -> acc[e] @ lane 24+e
// Means from a per-lane running sum + shfl_xor(16).
// TR=true: X/XP/OUT are [B][N][C] (contiguous gathers, b64 loads into
// even-aligned pairs feeding WMMA directly); else native [B][C][N].
// ---------------------------------------------------------------------------
template<bool TR>
__global__ __launch_bounds__(256) void ssim_edge_k(const float* __restrict__ X,
                                                   const float* __restrict__ XP,
                                                   const int* __restrict__ E,
                                                   float* __restrict__ OUT)
{
    const int lane = threadIdx.x & 31;
    const int w    = blockIdx.x * 8 + (threadIdx.x >> 5);   // (b,n) id, exact grid
    const int b = w / N_;
    const int n = w - b * N_;

    const float* Xb  = X  + (size_t)b * C_ * N_;
    const float* XPb = XP + (size_t)b * C_ * N_;
    const int* Ej = E + ((size_t)b * N_ + n) * K_;          // edge_index[0]: neighbors j
    const int* Ei = Ej + (size_t)B_ * N_ * K_;              // edge_index[1]: centers   i

    const int r    = lane & 15;   // row within 16-row WMMA group
    const int half = lane >> 4;   // 0 -> channels c0+0,1 ; 1 -> channels c0+2,3

    float o0 = 0.f, o1 = 0.f, o2 = 0.f;

    for (int g = 0; g < 3; ++g) {               // 3 groups of 8 edges (18 -> pad)
        int e = g * 8 + (r & 15 & 7);
        if (e > K_ - 1) e = K_ - 1;             // duplicate last edge; discarded later
        const int myIdx = (r < 8) ? Ei[e] : Ej[e];

        v8f acc = {};
        float s = 0.f;
#pragma unroll
        for (int c0 = 0; c0 < C_; c0 += 4) {
            const int c = c0 + half * 2;
            v2f a;
            if (TR) {
                // 8-byte aligned (row stride 384B, c even): single global_load_b64
                a = *(const v2f*)(Xb + (size_t)myIdx * C_ + c);
            } else {
                a[0] = Xb[(size_t)c * N_ + myIdx];
                a[1] = Xb[(size_t)(c + 1) * N_ + myIdx];
            }
            acc = __builtin_amdgcn_wmma_f32_16x16x4_f32(
                      false, a, false, a, (short)0, acc, false, false);
            s += a[0] + a[1];                   // running row-sum (half of channels)
        }
        s += __shfl_xor(s, 16, 32);             // full 96-channel row sum, all lanes

#pragma unroll
        for (int e8 = 0; e8 < 8; ++e8) {
            const int k = g * 8 + e8;
            if (k < K_) {                       // wave-uniform; drops padded edges
                const float inv = 1.0f / (float)C_;
                float ii = __shfl(acc[e8], e8,      32);
                float ij = __shfl(acc[e8], 8  + e8, 32);
                float jj = __shfl(acc[e8], 24 + e8, 32);
                float mi = __shfl(s, e8,     32) * inv;
                float mj = __shfl(s, 8 + e8, 32) * inv;
                float ivar = ii * inv - mi * mi;
                float jvar = jj * inv - mj * mj;
                float cov  = ij * inv - mi * mj;
                float num = (2.f * mi * mj + 1e-6f) * (2.f * cov + 1e-6f);
                float den = (mi * mi + mj * mj + 1e-6f) * (ivar + jvar + 1e-6f);
                float sff = 1.f - num / den;

                int ki = __shfl(myIdx, e8,     32);
                int kj = __shfl(myIdx, 8 + e8, 32);
#pragma unroll
                for (int t = 0; t < 3; ++t) {   // each lane owns 3 channels
                    const int c = lane + 32 * t;
                    float pi, pj;
                    if (TR) { pi = XPb[(size_t)ki * C_ + c]; pj = XPb[(size_t)kj * C_ + c]; }
                    else    { pi = XPb[(size_t)c * N_ + ki]; pj = XPb[(size_t)c * N_ + kj]; }
                    float v = pi + pj + fabsf(pi - pj) * sff;
                    if (t == 0) o0 += v; else if (t == 1) o1 += v; else o2 += v;
                }
            }
        }
    }

    if (TR) {
        float* o = OUT + ((size_t)b * N_ + n) * C_;         // coalesced
        o[lane] = o0; o[lane + 32] = o1; o[lane + 64] = o2;
    } else {
        OUT[((size_t)b * C_ + lane)      * N_ + n] = o0;
        OUT[((size_t)b * C_ + lane + 32) * N_ + n] = o1;
        OUT[((size_t)b * C_ + lane + 64) * N_ + n] = o2;
    }
}

// ---------------------------------------------------------------------------
extern "C" void kernel_launch(void* const* d_in, const int* in_sizes, int n_in,
                              void* d_out, int out_size, void* d_ws, size_t ws_size,
                              hipStream_t stream)
{
    const float* x  = (const float*)d_in[0];
    const float* xp = (const float*)d_in[1];
    const int*   e  = (const int*)d_in[2];
    float* out = (float*)d_out;

    const size_t tensorElems = (size_t)B_ * C_ * N_;        // 2.41M floats (~9.6 MB)
    const size_t tensorBytes = tensorElems * sizeof(float);
    const int nBlocks = (B_ * N_) / 8;                      // 25088 waves / 8 per block

    if (ws_size >= 3 * tensorBytes) {
        float* xt  = (float*)d_ws;
        float* xpt = xt  + tensorElems;
        float* ot  = xpt + tensorElems;
        dim3 tb(32, 8);
        // [B][C][N] -> [B][N][C]
        transpose_k<<<dim3((N_ + 31) / 32, (C_ + 31) / 32, B_), tb, 0, stream>>>(x,  xt,  C_, N_);
        transpose_k<<<dim3((N_ + 31) / 32, (C_ + 31) / 32, B_), tb, 0, stream>>>(xp, xpt, C_, N_);
        ssim_edge_k<true><<<nBlocks, 256, 0, stream>>>(xt, xpt, e, ot);
        // [B][N][C] -> [B][C][N]
        transpose_k<<<dim3((C_ + 31) / 32, (N_ + 31) / 32, B_), tb, 0, stream>>>(ot, out, N_, C_);
    } else {
        // Fallback: identical math, strided gathers on native layout.
        ssim_edge_k<false><<<nBlocks, 256, 0, stream>>>(x, xp, e, out);
    }
}